// dynamicGRU_39118562132046
// MI455X (gfx1250) — compile-verified
//
#include <hip/hip_runtime.h>
#include <hip/hip_bf16.h>

typedef __attribute__((ext_vector_type(2))) float v2f;
typedef __attribute__((ext_vector_type(8))) float v8f;

namespace {
constexpr int Bsz = 256;   // batch
constexpr int BEH = 4;     // behaviors
constexpr int Lt  = 50;    // time steps
constexpr int Id  = 128;   // input dim
constexpr int Cc  = 64;    // capsules
constexpr int Dd  = 16;    // capsule length
constexpr int N3  = 48;    // 3*D
constexpr int MT  = 16;    // batch rows per block (WMMA M)
constexpr int KT  = Id / 4;  // 32 k-tiles for the x GEMM
}

__device__ __forceinline__ v8f wmma4(v2f a, v2f b, v8f c) {
  // D = A(16x4,f32) * B(4x16,f32) + C(16x16,f32)
  return __builtin_amdgcn_wmma_f32_16x16x4_f32(
      /*neg_a=*/false, a, /*neg_b=*/false, b,
      /*c_mod=*/(short)0, c, /*reuse_a=*/false, /*reuse_b=*/false);
}

__device__ __forceinline__ float sigmoid_f(float v) {
  return 1.0f / (1.0f + __expf(-v));
}

__device__ __forceinline__ v8f splat8(float v) {
  v8f r;
#pragma unroll
  for (int i = 0; i < 8; ++i) r[i] = v;
  return r;
}

// One wave32 per block. Block owns: 16 batch rows (b0..b0+15), one behavior,
// one capsule. Runs the entire L=50 recurrence with H in registers.
__global__ __launch_bounds__(32) void dyn_gru_caps_kernel(
    const float* __restrict__ x,    // [B,BEH,L,I]
    const float* __restrict__ cin,  // [B,BEH,L,C,D]
    const float* __restrict__ Wi,   // [BEH,I,C,3D]
    const float* __restrict__ Wh,   // [BEH,C,D,3D]
    const float* __restrict__ Wc,   // [BEH,C,D,3D]
    const float* __restrict__ Bx,   // [BEH,C,3D]
    const float* __restrict__ Bh,   // [BEH,C,3D]
    const float* __restrict__ Bc,   // [BEH,C,3D]
    float* __restrict__ outH,       // [B,BEH,C,D]
    float* __restrict__ outN)       // [B,BEH,L,C,D]
{
  const int lane = threadIdx.x;         // 0..31
  const int ml   = lane & 15;           // row (A) / col (B,C)
  const int kh   = lane >> 4;           // k-half select
  const int cap  = blockIdx.x;          // capsule
  const int b0   = blockIdx.y * MT;     // batch tile base
  const int beh  = blockIdx.z;          // behavior

  // Wi slice [128 x 48] pre-swizzled to B-fragment order: [kk][j][lane][2]
  __shared__ __align__(16) float wiB[KT * 3 * 32 * 2];   // 24 KB
  __shared__ __align__(16) float hT[MT * Dd];            // 1 KB transpose staging

  // ---- stage Wi B-fragments into LDS (once, amortized over 50 steps) ----
#pragma unroll 4
  for (int kk = 0; kk < KT; ++kk) {
    const int k0 = 4 * kk + 2 * kh;
#pragma unroll
    for (int j = 0; j < 3; ++j) {
      const int n = 16 * j + ml;
      const long g = ((long)(beh * Id + k0) * Cc + cap) * N3 + n;
      const int idx = ((kk * 3 + j) * 32 + lane) * 2;
      wiB[idx]     = Wi[g];                      // k = k0
      wiB[idx + 1] = Wi[g + (long)Cc * N3];      // k = k0+1
    }
  }

  // ---- Wh / Wc B-fragments live in registers (16x48 each) ----
  v2f whB[4][3], wcB[4][3];
#pragma unroll
  for (int kk = 0; kk < 4; ++kk) {
    const int k0 = 4 * kk + 2 * kh;
#pragma unroll
    for (int j = 0; j < 3; ++j) {
      const int n = 16 * j + ml;
      const long g = ((long)(beh * Cc + cap) * Dd + k0) * N3 + n;
      v2f wc; wc[0] = Wc[g]; wc[1] = Wc[g + N3];
      v2f wh; wh[0] = Wh[g]; wh[1] = Wh[g + N3];
      wcB[kk][j] = wc;
      whB[kk][j] = wh;
    }
  }

  // ---- biases: C-layout value depends only on column (16*j + ml) ----
  float bx[3], bh[3], bc[3];
#pragma unroll
  for (int j = 0; j < 3; ++j) {
    const long g = ((long)beh * Cc + cap) * N3 + 16 * j + ml;
    bx[j] = Bx[g]; bh[j] = Bh[g]; bc[j] = Bc[g];
  }

  __syncthreads();

  // Row bases for streaming inputs (row = b0 + ml for A-fragments)
  const long xrow = ((long)(b0 + ml) * BEH + beh) * (long)Lt * Id;
  const long crow = ((long)(b0 + ml) * BEH + beh) * (long)Lt * Cc * Dd
                    + (long)cap * Dd;

  v8f H = splat8(0.0f);   // H0 = 0, C/D accumulator layout

#pragma unroll 1
  for (int l = 0; l < Lt; ++l) {
    const float* xp = x   + xrow + (long)l * Id;
    const float* cp = cin + crow + (long)l * (Cc * Dd);

    // hint the next timestep's input rows toward the caches
    if (l + 1 < Lt) {
      __builtin_prefetch(xp + Id, 0, 3);
      __builtin_prefetch(cp + Cc * Dd, 0, 3);
    }

    // ---- gi = x_l @ Wi[beh,:,cap,:] + Bx   (16x128 * 128x48) ----
    // unroll capped at 8: keeps 8 b64 loads in flight per burst while
    // holding peak VGPR live-set under the 256-register direct window.
    v8f gi0 = splat8(bx[0]), gi1 = splat8(bx[1]), gi2 = splat8(bx[2]);
#pragma unroll 8
    for (int kk = 0; kk < KT; ++kk) {
      v2f a = *(const v2f*)(xp + 4 * kk + 2 * kh);
      const int fb = (kk * 3) * 32 + lane;
      v2f f0 = *(const v2f*)&wiB[(fb)      * 2];
      v2f f1 = *(const v2f*)&wiB[(fb + 32) * 2];
      v2f f2 = *(const v2f*)&wiB[(fb + 64) * 2];
      gi0 = wmma4(a, f0, gi0);
      gi1 = wmma4(a, f1, gi1);
      gi2 = wmma4(a, f2, gi2);
    }

    // ---- gc = c_l @ Wc[beh,cap] + Bc   (16x16 * 16x48) ----
    v8f gc0 = splat8(bc[0]), gc1 = splat8(bc[1]), gc2 = splat8(bc[2]);
#pragma unroll
    for (int kk = 0; kk < 4; ++kk) {
      v2f a = *(const v2f*)(cp + 4 * kk + 2 * kh);
      gc0 = wmma4(a, wcB[kk][0], gc0);
      gc1 = wmma4(a, wcB[kk][1], gc1);
      gc2 = wmma4(a, wcB[kk][2], gc2);
    }

    // ---- re-layout H (C/D layout) -> A-fragment layout via LDS ----
#pragma unroll
    for (int r = 0; r < 8; ++r)
      hT[(r + 8 * kh) * Dd + ml] = H[r];
    __syncthreads();

    // ---- gh = H @ Wh[beh,cap] + Bh   (16x16 * 16x48) ----
    v8f gh0 = splat8(bh[0]), gh1 = splat8(bh[1]), gh2 = splat8(bh[2]);
#pragma unroll
    for (int kk = 0; kk < 4; ++kk) {
      v2f a = *(const v2f*)&hT[ml * Dd + 4 * kk + 2 * kh];
      gh0 = wmma4(a, whB[kk][0], gh0);
      gh1 = wmma4(a, whB[kk][1], gh1);
      gh2 = wmma4(a, whB[kk][2], gh2);
    }
    __syncthreads();

    // ---- gates + state update (elementwise on C/D layout) ----
    v8f Nt;
#pragma unroll
    for (int e = 0; e < 8; ++e) {
      const float R  = sigmoid_f(gi0[e] * gc0[e] + gh0[e]);
      const float Z  = sigmoid_f(gi1[e] * gc1[e] + gh1[e]);
      const float Nv = tanhf(gi2[e] * gc2[e] + R * gh2[e]);
      H[e]  = Z * Nv + (1.0f - Z) * H[e];
      Nt[e] = Nv;
    }

    // ---- store N_l: row m = r + 8*kh, col d = ml ----
#pragma unroll
    for (int r = 0; r < 8; ++r) {
      const long b = b0 + r + 8 * kh;
      outN[((((b * BEH + beh) * Lt + l) * Cc + cap) * (long)Dd) + ml] = Nt[r];
    }
  }

  // ---- store Hlast ----
#pragma unroll
  for (int r = 0; r < 8; ++r) {
    const long b = b0 + r + 8 * kh;
    outH[(((b * BEH + beh) * Cc + cap) * (long)Dd) + ml] = H[r];
  }
}

extern "C" void kernel_launch(void* const* d_in, const int* in_sizes, int n_in,
                              void* d_out, int out_size, void* d_ws, size_t ws_size,
                              hipStream_t stream) {
  (void)in_sizes; (void)n_in; (void)out_size; (void)d_ws; (void)ws_size;
  const float* x  = (const float*)d_in[0];
  const float* c  = (const float*)d_in[1];
  const float* Wi = (const float*)d_in[2];
  const float* Wh = (const float*)d_in[3];
  const float* Wc = (const float*)d_in[4];
  const float* Bx = (const float*)d_in[5];
  const float* Bh = (const float*)d_in[6];
  const float* Bc = (const float*)d_in[7];

  float* outH = (float*)d_out;                               // [B,BEH,C,D]
  float* outN = outH + (long)Bsz * BEH * Cc * Dd;            // [B,BEH,L,C,D]

  dim3 grid(Cc, Bsz / MT, BEH);   // (64, 16, 4) blocks, 1 wave32 each
  dim3 block(32);
  hipLaunchKernelGGL(dyn_gru_caps_kernel, grid, block, 0, stream,
                     x, c, Wi, Wh, Wc, Bx, Bh, Bc, outH, outN);
}